// BoundsChecker_82420422410955
// MI455X (gfx1250) — compile-verified
//
#include <hip/hip_runtime.h>
#include <math.h>

// ---------------- problem constants (match reference) ----------------
#define NREF   10000
#define NPAD   10240              // 16*640 padded candidates; index fits in 14 bits
#define NTILES (NPAD / 16)        // 640
#define NSPLIT 4                  // candidate-range splits per 16-query tile
#define TILES_PER_SPLIT (NTILES / NSPLIT)   // 160
#define DR_F   0.5f
#define L_F    5000.0f            // N * DR
#define BTRAJ  4096
#define GORD   7
#define MQ     (BTRAJ * GORD)     // 28672 query points (16 | MQ)
#define QTILES (MQ / 16)          // 1792
#define EPS_F        1e-4f
#define DELTA_EPS_F  0.01f
#define MAX_STEP_F   1.0f
#define STDEV_FACTOR 0.56568542494923806f   // 1/(sqrt(2)*1.25)
#define D2_BIAS  16.0f            // keeps WMMA d2 strictly positive despite f32 cancellation
#define IDX_MASK 0x3FFFu          // low 14 bits: candidate index
#define KEY_MASK 0xFFFFC000u      // high 18 bits: f32 bits of biased d2 (>=0)

// d_out layout (flat, reference tuple order)
#define OFF_R    0
#define OFF_CPV  28672
#define OFF_CPT  86016
#define OFF_CPN  143360
#define OFF_DLT  200704
#define OFF_SD   258048
#define OFF_LW   286720
#define OFF_RW   315392
#define OFF_PL   344064
#define OFF_PR   372736
#define OFF_NOL  401408
#define OFF_NOR  405504

// workspace layout
#define WS_KEYS_OFF (NPAD * 4 * sizeof(float))   // 163840 B of B-matrix, then keys

typedef float v2f __attribute__((ext_vector_type(2)));
typedef float v8f __attribute__((ext_vector_type(8)));

// ---------- phase 0: pack WMMA B-matrix + reset key buffer ----------
// B column n = [Px, Py, |P|^2, 1]; with A row = [-2x, -2y, 1, |x|^2+16] the WMMA
// emits the full biased squared distance d2+16 (> 0, so u32 ordering == f32 ordering).
__global__ void bc_init(const float* __restrict__ P, float* __restrict__ Bws,
                        unsigned int* __restrict__ keys) {
    const int i = blockIdx.x * blockDim.x + threadIdx.x;
    if (i < NPAD) {
        float px = 0.0f, py = 0.0f, q = 1.0e30f;   // padding never wins the argmin
        if (i < NREF) { px = P[2*i]; py = P[2*i+1]; q = px*px + py*py; }
        reinterpret_cast<float4*>(Bws)[i] = make_float4(px, py, q, 1.0f);
    }
    if (i < MQ) keys[i] = 0xFFFFFFFFu;
}

// ---------- phase 1: WMMA nearest-sample argmin (packed-key min) ----------
// 4 waves cooperate on each 16-query tile, each scanning 1/4 of the candidates;
// partial argmins merge via global_atomic_min_u32 on packed (d2|index) keys.
__global__ void __launch_bounds__(256) bc_nn_search(const float* __restrict__ pos,
                                                    const float* __restrict__ Bws,
                                                    unsigned int* __restrict__ keys) {
    const int lane  = threadIdx.x & 31;
    const int wave  = blockIdx.x * (blockDim.x >> 5) + (threadIdx.x >> 5);
    const int qtile = wave >> 2;                 // NSPLIT = 4
    const int split = wave & (NSPLIT - 1);
    const int mBase = qtile * 16;
    const int sub   = lane & 15;
    const bool hi   = lane >= 16;

    // A operand (16x4 f32): lanes 0-15 -> K=0,1 = (-2x,-2y); lanes 16-31 -> K=2,3 = (1, |x|^2+16)
    v2f a;
    {
        const int m = mBase + sub;
        const float x = pos[2*m], y = pos[2*m+1];
        a.x = hi ? 1.0f : -2.0f * x;
        a.y = hi ? (x*x + y*y + D2_BIAS) : -2.0f * y;
    }

    unsigned int mink[8];
#pragma unroll
    for (int v = 0; v < 8; ++v) mink[v] = 0xFFFFFFFFu;

    // B operand per lane: col = n + sub; lanes 0-15 read (Px,Py), lanes 16-31 read (q,1)
    const int nStart = split * TILES_PER_SPLIT * 16;
    const float* p = Bws + nStart * 4 + sub * 4 + (hi ? 2 : 0);
    int nlane = nStart + sub;
    for (int t = 0; t < TILES_PER_SPLIT; ++t) {
        __builtin_prefetch(p + 512, 0, 0);       // global_prefetch_b8, 8 tiles ahead
        v2f b; b.x = p[0]; b.y = p[1];           // merged global_load_b64
        v8f c = {0.f, 0.f, 0.f, 0.f, 0.f, 0.f, 0.f, 0.f};
        c = __builtin_amdgcn_wmma_f32_16x16x4_f32(false, a, false, b,
                                                  (short)0, c, false, false);
#pragma unroll
        for (int v = 0; v < 8; ++v) {            // v_and_or_b32 + v_min_u32 per row
            const unsigned int kb = (__float_as_uint(c[v]) & KEY_MASK) | (unsigned int)nlane;
            mink[v] = kb < mink[v] ? kb : mink[v];
        }
        nlane += 16;
        p += 64;
    }

    // butterfly min across the 16 lanes of each half (wave32: masks 1,2,4,8)
#pragma unroll
    for (int offs = 1; offs <= 8; offs <<= 1) {
#pragma unroll
        for (int v = 0; v < 8; ++v) {
            const unsigned int ok = (unsigned int)__shfl_xor((int)mink[v], offs, 32);
            mink[v] = ok < mink[v] ? ok : mink[v];
        }
    }
    // merge the 4 candidate-range splits
    if (lane == 0) {
#pragma unroll
        for (int v = 0; v < 8; ++v) atomicMin(&keys[mBase + v], mink[v]);       // rows 0-7
    } else if (lane == 16) {
#pragma unroll
        for (int v = 0; v < 8; ++v) atomicMin(&keys[mBase + 8 + v], mink[v]);   // rows 8-15
    }
}

// ---------- phase 2: Newton refine + per-query outputs ----------
__device__ __forceinline__ void bc_eval_curve(float r, const float* __restrict__ P,
                                              float2& val, float2& tang, float2& nrm) {
    const float u  = r * (1.0f / DR_F);     // r in [0, L)
    const float fu = floorf(u);
    int i0 = (int)fu; if (i0 >= NREF) i0 -= NREF;
    int i1 = i0 + 1;  if (i1 >= NREF) i1 = 0;
    const float frac = u - fu;
    const float p0x = P[2*i0], p0y = P[2*i0+1];
    const float p1x = P[2*i1], p1y = P[2*i1+1];
    val.x = p0x * (1.0f - frac) + p1x * frac;
    val.y = p0y * (1.0f - frac) + p1y * frac;
    const float sx = (p1x - p0x) * (1.0f / DR_F);
    const float sy = (p1y - p0y) * (1.0f / DR_F);
    const float inv = 1.0f / sqrtf(sx*sx + sy*sy);
    tang.x = sx * inv; tang.y = sy * inv;
    nrm.x = -tang.y;   nrm.y = tang.x;
}

__device__ __forceinline__ float bc_erf_pos(float x) {   // x >= 0, A&S 7.1.26
    const float t = 1.0f / (1.0f + 0.3275911f * x);
    const float y = t * (0.254829592f + t * (-0.284496736f + t * (1.421413741f +
                     t * (-1.453152027f + t * 1.061405429f))));
    return 1.0f - y * expf(-x * x);
}

__global__ void bc_refine(const float* __restrict__ pos, const float* __restrict__ P,
                          const float* __restrict__ arcl, const float* __restrict__ lwid,
                          const float* __restrict__ rwid, const int* __restrict__ niterp,
                          const unsigned int* __restrict__ keys, float* __restrict__ out) {
    const int m = blockIdx.x * blockDim.x + threadIdx.x;
    if (m >= MQ) return;
    const float px = pos[2*m], py = pos[2*m+1];
    float r = arcl[keys[m] & IDX_MASK];
    const int niter = *niterp;
    for (int it = 0; it < niter; ++it) {
        float2 val, tang, nrm;
        bc_eval_curve(r, P, val, tang, nrm);
        const float g = (px - val.x) * tang.x + (py - val.y) * tang.y;
        const float step = fminf(fmaxf(g, -MAX_STEP_F), MAX_STEP_F);
        const bool active = (fabsf(g) > EPS_F) && (fabsf(step) > DELTA_EPS_F);
        r = r + (active ? step : 0.0f);
        r = fmodf(r, L_F);
        if (r < 0.0f) r += L_F;
    }
    float2 val, tang, nrm;
    bc_eval_curve(r, P, val, tang, nrm);
    const float dx = px - val.x, dy = py - val.y;
    const float sd = dx * nrm.x + dy * nrm.y;

    // jnp.interp (uniform knots, clamp at last knot, no wrap)
    const float u  = r * (1.0f / DR_F);
    const float fu = floorf(u);
    const int   i0 = (int)fu;
    const float fr = u - fu;
    float lwv, rwv;
    if (i0 >= NREF - 1) { lwv = lwid[NREF-1]; rwv = rwid[NREF-1]; }
    else {
        lwv = lwid[i0] + (lwid[i0+1] - lwid[i0]) * fr;
        rwv = rwid[i0] + (rwid[i0+1] - rwid[i0]) * fr;
    }
    const float pl = bc_erf_pos(fmaxf(sd - lwv, 0.0f) * STDEV_FACTOR);
    const float pr = bc_erf_pos(fmaxf(rwv - sd, 0.0f) * STDEV_FACTOR);

    out[OFF_R   + m]       = r;
    out[OFF_CPV + 2*m]     = val.x;  out[OFF_CPV + 2*m + 1] = val.y;
    out[OFF_CPT + 2*m]     = tang.x; out[OFF_CPT + 2*m + 1] = tang.y;
    out[OFF_CPN + 2*m]     = nrm.x;  out[OFF_CPN + 2*m + 1] = nrm.y;
    out[OFF_DLT + 2*m]     = dx;     out[OFF_DLT + 2*m + 1] = dy;
    out[OFF_SD  + m]       = sd;
    out[OFF_LW  + m]       = lwv;
    out[OFF_RW  + m]       = rwv;
    out[OFF_PL  + m]       = pl;
    out[OFF_PR  + m]       = pr;
}

// ---------- phase 3: Gauss-Legendre quadrature + survival prob ----------
__global__ void bc_quad(float* __restrict__ out) {
    const int b = blockIdx.x * blockDim.x + threadIdx.x;
    if (b >= BTRAJ) return;
    // leggauss(7) weights scaled by 0.5*DT = 1.5
    const float w[7] = {0.19422744925330455f, 0.41955808723391490f, 0.57274507575767840f,
                        0.62693877551020410f, 0.57274507575767840f, 0.41955808723391490f,
                        0.19422744925330455f};
    float ll = 0.0f, lr = 0.0f;
#pragma unroll
    for (int g = 0; g < GORD; ++g) {
        ll += out[OFF_PL + b * GORD + g] * w[g];
        lr += out[OFF_PR + b * GORD + g] * w[g];
    }
    out[OFF_NOL + b] = expf(-ll);
    out[OFF_NOR + b] = expf(-lr);
}

// ---------------- launcher ----------------
extern "C" void kernel_launch(void* const* d_in, const int* in_sizes, int n_in,
                              void* d_out, int out_size, void* d_ws, size_t ws_size,
                              hipStream_t stream) {
    const float* positions = (const float*)d_in[0];
    const float* refline   = (const float*)d_in[1];
    const float* arcl      = (const float*)d_in[2];
    const float* lwid      = (const float*)d_in[3];
    const float* rwid      = (const float*)d_in[4];
    const int*   niter     = (const int*)  d_in[5];
    float* out = (float*)d_out;
    float* Bws = (float*)d_ws;
    unsigned int* keys = (unsigned int*)((char*)d_ws + WS_KEYS_OFF);

    bc_init<<<(MQ + 255) / 256, 256, 0, stream>>>(refline, Bws, keys);
    bc_nn_search<<<(QTILES * NSPLIT) / 8, 256, 0, stream>>>(positions, Bws, keys); // 8 waves/WG
    bc_refine<<<(MQ + 255) / 256, 256, 0, stream>>>(positions, refline, arcl, lwid, rwid,
                                                    niter, keys, out);
    bc_quad<<<(BTRAJ + 255) / 256, 256, 0, stream>>>(out);
}